// Attention_81269371175472
// MI455X (gfx1250) — compile-verified
//
#include <hip/hip_runtime.h>
#include <hip/hip_bf16.h>

// ---------------- problem constants ----------------
#define Bb   32
#define Tt   4
#define Dd   2048
#define Hh   16
#define Gg   4
#define Kk   128
#define Ss   4096
#define NREP 4            // H / G
#define EPS  1e-6f
#define ROPE_BASE 10000.0f

#define NCHUNK 8          // S split for flash-decoding
#define CHUNK  (Ss / NCHUNK)
#define NPART  (NCHUNK + 1)          // +1 for the "fresh" (uncached) keys
#define PSTRIDE (16 * (Kk + 2))      // per-partial floats: m[16], l[16], acc[16][128]
#define MASKVAL (-3.0e38f)
#define QK_SCALE 0.08838834764831845f   // 1/sqrt(128)

typedef float v2f __attribute__((ext_vector_type(2)));
typedef float v8f __attribute__((ext_vector_type(8)));
typedef unsigned int u32x4 __attribute__((ext_vector_type(4)));
typedef int i32x4 __attribute__((ext_vector_type(4)));
typedef int i32x8 __attribute__((ext_vector_type(8)));

// =====================================================================
// Tensor Data Mover descriptor helpers (CDNA5 D#, CDNA5 ISA)
// 2-D tile: tile_w elements/row (f32), tile_h rows, global row stride in
// elements. TDM LDS padding inserts 4 dwords after every 2^(iv+1) dwords,
// so LDS row stride = row dwords + 4 (bank-conflict-free frag reads).
// =====================================================================
__device__ __forceinline__ i32x8 tdm_make_g1(unsigned tile_w, unsigned tile_h,
                                             unsigned pad_iv_code,
                                             unsigned long long row_stride_elems)
{
    unsigned d0 = (2u << 16)               // data_size = 4 bytes
                | (1u << 20)               // pad_enable (load only)
                | (pad_iv_code << 22)      // pad_interval: 2^(code+1) dwords
                | (3u << 25);              // pad_amount: 4 dwords
    unsigned long long tdim0 = 0x7FFFFFFFull;   // no OOB clipping
    unsigned long long tdim1 = 0x7FFFFFFFull;
    unsigned d1 = (unsigned)((tdim0 & 0xFFFFull) << 16);
    unsigned d2 = (unsigned)((tdim0 >> 16) & 0xFFFFull)
                | (unsigned)((tdim1 & 0xFFFFull) << 16);
    unsigned d3 = (unsigned)((tdim1 >> 16) & 0xFFFFull)
                | ((tile_w & 0xFFFFu) << 16);          // tile_dim0
    unsigned d4 = (tile_h & 0xFFFFu);                  // tile_dim1 (tile_dim2=0)
    unsigned d5 = (unsigned)(row_stride_elems & 0xFFFFFFFFull);  // dim0_stride lo
    unsigned d6 = (unsigned)((row_stride_elems >> 32) & 0xFFFFull); // hi16
    unsigned d7 = 0;
    i32x8 g;
    g[0]=(int)d0; g[1]=(int)d1; g[2]=(int)d2; g[3]=(int)d3;
    g[4]=(int)d4; g[5]=(int)d5; g[6]=(int)d6; g[7]=(int)d7;
    return g;
}

__device__ __forceinline__ void tdm_load(void* ldsPtr, const void* gPtr, i32x8 g1)
{
    unsigned long long ga = (unsigned long long)(uintptr_t)gPtr;
    u32x4 g0;
    g0[0] = 1u;                                          // count=1, user mode
    g0[1] = (unsigned)(uintptr_t)ldsPtr;                 // LDS byte address
    g0[2] = (unsigned)(ga & 0xFFFFFFFFull);              // global addr lo
    g0[3] = (unsigned)((ga >> 32) & 0x01FFFFFFull) | (2u << 30); // hi + type=2
    i32x4 z4 = (i32x4){0, 0, 0, 0};
    i32x8 z8 = (i32x8){0, 0, 0, 0, 0, 0, 0, 0};
    // 6-arg form (clang-23 / therock-10.0): (g0, g1, g2, g3, g4, cpol)
    __builtin_amdgcn_tensor_load_to_lds(g0, g1, z4, z4, z8, 0);
}

// =====================================================================
// fp32 GEMM C = A(MxKd) * B(KdxN); one wave -> 16x64 tile.
// TDM double-buffers 16-deep K-blocks of A and B into LDS; 16 WMMA
// f32 16x16x4 per K-block overlap the next block's DMA.
// =====================================================================
__global__ __launch_bounds__(32)
void gemm_f32_wmma(const float* __restrict__ A, const float* __restrict__ Bw,
                   float* __restrict__ C, int M, int N, int Kd)
{
    __shared__ float As[2][16 * 20];   // 16 rows, padded stride 20
    __shared__ float Bs[2][16 * 68];   // 16 rows, padded stride 68

    const int lane = threadIdx.x;
    const int mt   = blockIdx.y * 16;
    const int nt   = blockIdx.x * 64;
    const int mrow = lane & 15;
    const int kb   = (lane >> 4) * 2;
    const int hi   = lane >> 4;

    const i32x8 g1A = tdm_make_g1(16, 16, 3, (unsigned long long)Kd); // pad every 16 dw
    const i32x8 g1B = tdm_make_g1(64, 16, 5, (unsigned long long)N);  // pad every 64 dw

    v8f acc[4];
    #pragma unroll
    for (int t4 = 0; t4 < 4; ++t4) acc[t4] = (v8f){0,0,0,0,0,0,0,0};

    const int nkb = Kd >> 4;

    tdm_load(&As[0][0], A + (size_t)mt * Kd, g1A);
    tdm_load(&Bs[0][0], Bw + nt, g1B);

    for (int kb2 = 0; kb2 < nkb; ++kb2) {
        const int bufc = kb2 & 1;
        if (kb2 + 1 < nkb) {
            const int k0n = (kb2 + 1) << 4;
            tdm_load(&As[bufc ^ 1][0], A + (size_t)mt * Kd + k0n, g1A);
            tdm_load(&Bs[bufc ^ 1][0], Bw + (size_t)k0n * N + nt, g1B);
            __builtin_amdgcn_s_wait_tensorcnt(2);   // current buffer's 2 DMAs done
        } else {
            __builtin_amdgcn_s_wait_tensorcnt(0);
        }
        const float* as = &As[bufc][0];
        const float* bs = &Bs[bufc][0];
        #pragma unroll
        for (int ks4 = 0; ks4 < 4; ++ks4) {
            const int kc = ks4 * 4 + kb;
            v2f a; a[0] = as[mrow * 20 + kc]; a[1] = as[mrow * 20 + kc + 1];
            #pragma unroll
            for (int t4 = 0; t4 < 4; ++t4) {
                v2f b;
                b[0] = bs[kc * 68 + t4 * 16 + mrow];
                b[1] = bs[(kc + 1) * 68 + t4 * 16 + mrow];
                acc[t4] = __builtin_amdgcn_wmma_f32_16x16x4_f32(false, a, false, b, (short)0, acc[t4], false, false);
            }
        }
    }

    const int colL = lane & 15;
    #pragma unroll
    for (int i = 0; i < 8; ++i) {
        const int row = mt + hi * 8 + i;
        float* cp = C + (size_t)row * N + nt + colL;
        cp[0]  = acc[0][i];
        cp[16] = acc[1][i];
        cp[32] = acc[2][i];
        cp[48] = acc[3][i];
    }
}

// =====================================================================
// RMS-norm + scale + RoPE (in place on workspace q / k). q additionally
// folds in the 1/sqrt(K) logit scale. One block per (b, t, head).
// =====================================================================
__global__ __launch_bounds__(128)
void rms_rope(float* __restrict__ qws, float* __restrict__ kws,
              const float* __restrict__ q_scale, const float* __restrict__ k_scale,
              const int* __restrict__ seg, const int* __restrict__ curp)
{
    const int nheads = Hh + Gg;
    int blk = blockIdx.x;
    int b = blk / (Tt * nheads);
    int r = blk % (Tt * nheads);
    int t = r / nheads;
    int hd = r % nheads;
    const bool isq = hd < Hh;

    float* base = isq ? (qws + (((size_t)(b * Tt + t) * Hh + hd) * Kk))
                      : (kws + (((size_t)(b * Tt + t) * Gg + (hd - Hh)) * Kk));
    const float* sc = isq ? q_scale : k_scale;
    const int cur = *curp;

    int csum = 0;
    for (int tt = 0; tt <= t; ++tt) csum += (seg[b * Tt + tt] != 0) ? 1 : 0;
    const int pos = csum - 1 + cur;

    const int j = threadIdx.x;
    float x = base[j];

    __shared__ float red[4];
    float ssq = x * x;
    ssq += __shfl_xor(ssq, 1, 32);
    ssq += __shfl_xor(ssq, 2, 32);
    ssq += __shfl_xor(ssq, 4, 32);
    ssq += __shfl_xor(ssq, 8, 32);
    ssq += __shfl_xor(ssq, 16, 32);
    if ((j & 31) == 0) red[j >> 5] = ssq;
    __syncthreads();
    float var = (red[0] + red[1] + red[2] + red[3]) * (1.0f / Kk);
    float xn = x * rsqrtf(var + EPS) * sc[j];

    __shared__ float buf[128];
    buf[j] = xn;
    __syncthreads();

    const int i = j & 63;
    float ts = __powf(ROPE_BASE, (float)i * (2.0f / (float)Kk));
    float ang = (float)pos / ts;
    float s, c;
    __sincosf(ang, &s, &c);
    float out;
    if (j < 64) out = xn * c - buf[j + 64] * s;
    else        out = xn * c + buf[j - 64] * s;
    if (isq) out *= QK_SCALE;
    base[j] = out;
}

// =====================================================================
// Flash-decoding chunk kernel. One wave per (chunk, g, b); 16 query rows
// (4 reps x 4 timesteps). Cached chunks stream 16-key K/V tiles via TDM
// (double-buffered); the extra chunk (ch == NCHUNK) covers fresh keys
// [cur, cur+T) from workspace (inputs are never mutated).
// =====================================================================
__global__ __launch_bounds__(32)
void attn_chunk(const float* __restrict__ qws, const float* __restrict__ kws,
                const float* __restrict__ vws,
                const float* __restrict__ kcache, const float* __restrict__ vcache,
                const int* __restrict__ curp, float* __restrict__ part)
{
    const int lane = threadIdx.x;
    const int ch = blockIdx.x;
    const int g  = blockIdx.y;
    const int b  = blockIdx.z;
    const int cur = *curp;

    __shared__ float qs[16][Kk + 4];          // stride 132 -> conflict-free
    __shared__ float ks[2][16][Kk + 4];       // TDM double buffers
    __shared__ float vs[2][16][Kk + 4];
    __shared__ float ps[16][20];

    // stage the 16 query rows (row j = rep*T + tq)
    for (int idx = lane; idx < 16 * 32; idx += 32) {
        int j  = idx >> 5;
        int c4 = (idx & 31) * 4;
        int rep = j >> 2, tq = j & 3;
        const float4 v = *(const float4*)(qws +
            (((size_t)(b * Tt + tq) * Hh + (g * NREP + rep)) * Kk) + c4);
        qs[j][c4 + 0] = v.x; qs[j][c4 + 1] = v.y;
        qs[j][c4 + 2] = v.z; qs[j][c4 + 3] = v.w;
    }
    __syncthreads();

    float m_run[8], l_run[8];
    v8f acc[8];
    #pragma unroll
    for (int i = 0; i < 8; ++i) {
        m_run[i] = MASKVAL; l_run[i] = 0.0f;
        acc[i] = (v8f){0,0,0,0,0,0,0,0};
    }

    const int mrow = lane & 15;
    const int kb   = (lane >> 4) * 2;
    const int hi   = lane >> 4;
    const int sj   = lane & 15;

    const bool fresh = (ch == NCHUNK);
    const int s0 = ch * CHUNK;

    int ntiles;
    if (fresh) {
        ntiles = 1;
    } else {
        int avail = cur - s0;
        if (avail < 0) avail = 0;
        if (avail > CHUNK) avail = CHUNK;
        ntiles = (avail + 15) >> 4;
    }

    const i32x8 g1kv = tdm_make_g1(Kk, 16, 6, (unsigned long long)(Gg * Kk)); // pad every 128 dw

    if (ntiles > 0) {
        if (fresh) {
            // stage the 4 fresh rows (zero-fill the rest) with plain loads
            for (int idx = lane; idx < 16 * 32; idx += 32) {
                int r  = idx >> 5;
                int c4 = (idx & 31) * 4;
                float4 kv = make_float4(0.f, 0.f, 0.f, 0.f);
                float4 vv = make_float4(0.f, 0.f, 0.f, 0.f);
                if (r < Tt) {
                    kv = *(const float4*)(kws + (((size_t)(b * Tt + r) * Gg + g) * Kk) + c4);
                    vv = *(const float4*)(vws + (((size_t)(b * Tt + r) * Gg + g) * Kk) + c4);
                }
                ks[0][r][c4+0]=kv.x; ks[0][r][c4+1]=kv.y; ks[0][r][c4+2]=kv.z; ks[0][r][c4+3]=kv.w;
                vs[0][r][c4+0]=vv.x; vs[0][r][c4+1]=vv.y; vs[0][r][c4+2]=vv.z; vs[0][r][c4+3]=vv.w;
            }
            __syncthreads();
        } else {
            const size_t off0 = (((size_t)b * Ss + s0) * Gg + g) * Kk;
            tdm_load(&ks[0][0][0], kcache + off0, g1kv);
            tdm_load(&vs[0][0][0], vcache + off0, g1kv);
        }

        for (int tile = 0; tile < ntiles; ++tile) {
            const int bufc = tile & 1;
            const int sbase = s0 + tile * 16;
            int nvalid;
            if (fresh) {
                nvalid = Tt;
            } else {
                nvalid = cur - sbase;
                if (nvalid > 16) nvalid = 16;
                if (tile + 1 < ntiles) {
                    const size_t off = (((size_t)b * Ss + (sbase + 16)) * Gg + g) * Kk;
                    tdm_load(&ks[bufc ^ 1][0][0], kcache + off, g1kv);
                    tdm_load(&vs[bufc ^ 1][0][0], vcache + off, g1kv);
                    __builtin_amdgcn_s_wait_tensorcnt(2);
                } else {
                    __builtin_amdgcn_s_wait_tensorcnt(0);
                }
            }

            // logits: Q(16x128) . K(16x128)^T -> 32 WMMA f32 16x16x4 steps
            v8f c = {0,0,0,0,0,0,0,0};
            #pragma unroll
            for (int kk = 0; kk < 32; ++kk) {
                v2f a;  a[0]  = qs[mrow][kk * 4 + kb];       a[1]  = qs[mrow][kk * 4 + kb + 1];
                v2f bb; bb[0] = ks[bufc][sj][kk * 4 + kb];   bb[1] = ks[bufc][sj][kk * 4 + kb + 1];
                c = __builtin_amdgcn_wmma_f32_16x16x4_f32(false, a, false, bb, (short)0, c, false, false);
            }

            // online softmax (rows across 16-lane groups; butterfly reduce)
            const bool colv = (sj < nvalid);
            #pragma unroll
            for (int i = 0; i < 8; ++i) {
                const int ri = i + hi * 8;
                bool valid = colv;
                if (fresh) valid = valid && (sj <= (ri & 3));   // causal vs fresh keys
                float lv = valid ? c[i] : MASKVAL;

                float mx = lv;
                mx = fmaxf(mx, __shfl_xor(mx, 1, 32));
                mx = fmaxf(mx, __shfl_xor(mx, 2, 32));
                mx = fmaxf(mx, __shfl_xor(mx, 4, 32));
                mx = fmaxf(mx, __shfl_xor(mx, 8, 32));

                float m_new = fmaxf(m_run[i], mx);
                float alpha = __expf(m_run[i] - m_new);
                float p = valid ? __expf(lv - m_new) : 0.0f;

                float rs = p;
                rs += __shfl_xor(rs, 1, 32);
                rs += __shfl_xor(rs, 2, 32);
                rs += __shfl_xor(rs, 4, 32);
                rs += __shfl_xor(rs, 8, 32);

                l_run[i] = l_run[i] * alpha + rs;
                m_run[i] = m_new;
                #pragma unroll
                for (int t8 = 0; t8 < 8; ++t8) acc[t8][i] *= alpha;

                ps[ri][sj] = p;                 // transpose P through LDS
            }
            __syncthreads();

            // acc += P(16x16) @ V(16x128): 8 col-tiles x 4 k-steps of WMMA
            #pragma unroll
            for (int t8 = 0; t8 < 8; ++t8) {
                #pragma unroll
                for (int kk = 0; kk < 4; ++kk) {
                    v2f a;  a[0]  = ps[mrow][kk * 4 + kb];   a[1]  = ps[mrow][kk * 4 + kb + 1];
                    v2f bb; bb[0] = vs[bufc][kk * 4 + kb][t8 * 16 + sj];
                            bb[1] = vs[bufc][kk * 4 + kb + 1][t8 * 16 + sj];
                    acc[t8] = __builtin_amdgcn_wmma_f32_16x16x4_f32(false, a, false, bb, (short)0, acc[t8], false, false);
                }
            }
            __syncthreads();   // ps reused next tile
        }
    }

    // write partials: [bg][ch] -> m[16], l[16], acc[16][128]
    const int bg = b * Gg + g;
    float* pb = part + ((size_t)bg * NPART + ch) * PSTRIDE;
    if (sj == 0) {
        #pragma unroll
        for (int i = 0; i < 8; ++i) {
            pb[hi * 8 + i]      = m_run[i];
            pb[16 + hi * 8 + i] = l_run[i];
        }
    }
    #pragma unroll
    for (int i = 0; i < 8; ++i) {
        const int ri = i + hi * 8;
        #pragma unroll
        for (int t8 = 0; t8 < 8; ++t8)
            pb[32 + ri * Kk + t8 * 16 + sj] = acc[t8][i];
    }
}

// =====================================================================
// Merge the NPART flash partials per (b, g); emit attn (B,T,H,K) to ws.
// =====================================================================
__global__ __launch_bounds__(128)
void attn_combine(const float* __restrict__ part, float* __restrict__ attn)
{
    const int bg = blockIdx.x;
    const int b = bg / Gg;
    const int g = bg % Gg;
    const int k = threadIdx.x;

    for (int j = 0; j < 16; ++j) {
        float m_tot = MASKVAL;
        for (int ch = 0; ch < NPART; ++ch)
            m_tot = fmaxf(m_tot, part[((size_t)bg * NPART + ch) * PSTRIDE + j]);
        float l_tot = 0.0f, a = 0.0f;
        for (int ch = 0; ch < NPART; ++ch) {
            const float* pb = part + ((size_t)bg * NPART + ch) * PSTRIDE;
            float w = __expf(pb[j] - m_tot);
            l_tot += w * pb[16 + j];
            a     += w * pb[32 + j * Kk + k];
        }
        const int rep = j >> 2, tq = j & 3;
        attn[(((size_t)(b * Tt + tq) * Hh) + g * NREP + rep) * Kk + k] = a / l_tot;
    }
}

// =====================================================================
// launch
// =====================================================================
extern "C" void kernel_launch(void* const* d_in, const int* in_sizes, int n_in,
                              void* d_out, int out_size, void* d_ws, size_t ws_size,
                              hipStream_t stream)
{
    const float* hidden = (const float*)d_in[0];
    const int*   seg    = (const int*)  d_in[1];
    const float* kcache = (const float*)d_in[2];
    const float* vcache = (const float*)d_in[3];
    const float* Wq     = (const float*)d_in[4];
    const float* Wk     = (const float*)d_in[5];
    const float* Wv     = (const float*)d_in[6];
    const float* Wo     = (const float*)d_in[7];
    const float* q_sc   = (const float*)d_in[8];
    const float* k_sc   = (const float*)d_in[9];
    const int*   curp   = (const int*)  d_in[10];

    float* ws      = (float*)d_ws;
    float* q_ws    = ws;                                   // B*T*H*K
    float* k_ws    = q_ws    + (size_t)Bb * Tt * Hh * Kk;  // B*T*G*K
    float* v_ws    = k_ws    + (size_t)Bb * Tt * Gg * Kk;  // B*T*G*K
    float* attn_ws = v_ws    + (size_t)Bb * Tt * Gg * Kk;  // B*T*H*K
    float* part_ws = attn_ws + (size_t)Bb * Tt * Hh * Kk;  // B*G*NPART*PSTRIDE

    const int M = Bb * Tt;   // 128

    // 1) QKV projections
    gemm_f32_wmma<<<dim3((Hh * Kk) / 64, M / 16), 32, 0, stream>>>(hidden, Wq, q_ws, M, Hh * Kk, Dd);
    gemm_f32_wmma<<<dim3((Gg * Kk) / 64, M / 16), 32, 0, stream>>>(hidden, Wk, k_ws, M, Gg * Kk, Dd);
    gemm_f32_wmma<<<dim3((Gg * Kk) / 64, M / 16), 32, 0, stream>>>(hidden, Wv, v_ws, M, Gg * Kk, Dd);

    // 2) RMS-norm + RoPE (q folds 1/sqrt(K))
    rms_rope<<<Bb * Tt * (Hh + Gg), 128, 0, stream>>>(q_ws, k_ws, q_sc, k_sc, seg, curp);

    // 3) flash-decoding chunks (cached S-splits + fresh keys)
    attn_chunk<<<dim3(NPART, Gg, Bb), 32, 0, stream>>>(q_ws, k_ws, v_ws, kcache, vcache, curp, part_ws);

    // 4) combine partials
    attn_combine<<<Bb * Gg, 128, 0, stream>>>(part_ws, attn_ws);

    // 5) output projection
    gemm_f32_wmma<<<dim3(Dd / 64, M / 16), 32, 0, stream>>>(attn_ws, Wo, (float*)d_out, M, Dd, Hh * Kk);
}